// HeteroGNN_10574209483360
// MI455X (gfx1250) — compile-verified
//
#include <hip/hip_runtime.h>
#include <hip/hip_bf16.h>

typedef float v2f __attribute__((ext_vector_type(2)));
typedef float v8f __attribute__((ext_vector_type(8)));

// ---------------------------------------------------------------------------
// fp32 WMMA GEMM:  C[M,NC] = A[M,K] @ B[K,NC]   (row-major, no bias)
// Compile-time K/NC -> all loads use immediate offsets (no per-iter addr math).
// Each wave computes a 16 x (16*NACC) tile: NACC accumulator chains share one
// A fragment (V_WMMA_F32_16X16X4_F32, A=16x4 across lanes per ISA layout).
// Block = 128 threads (4 waves); block tile = 16 rows x (64*NACC) cols = NC.
// ---------------------------------------------------------------------------
template <int K, int NC, int NACC>
__global__ __launch_bounds__(128) void gemm16_wmma(
    const float* __restrict__ A, const float* __restrict__ B,
    float* __restrict__ C, int M) {
  const unsigned lane = threadIdx.x & 31u;
  const unsigned wave = threadIdx.x >> 5;            // 0..3
  const int row0 = blockIdx.x * 16;
  const int col0 = (int)wave * (16 * NACC);          // wave-uniform, 4*16*NACC==NC
  if (row0 >= M) return;                             // wave-uniform exit

  const int ml = (int)(lane & 15u);                  // M (for A) / N (for B)
  const int kh = (int)((lane >> 4u) << 1u);          // 0 or 2 (K sub-phase)
  const int rowL = (row0 + ml < M) ? (row0 + ml) : (M - 1);
  const float* __restrict__ arow = A + (size_t)rowL * K + kh;
  const float* __restrict__ bp   = B + (size_t)kh * NC + col0 + ml;

  v8f acc[NACC];
#pragma unroll
  for (int t = 0; t < NACC; ++t) acc[t] = (v8f){};

#pragma unroll 1
  for (int k0 = 0; k0 < K; k0 += 16) {
    __builtin_prefetch(arow + k0 + 64, 0, 1);        // next A panel (global_prefetch)
#pragma unroll
    for (int kk = 0; kk < 16; kk += 4) {
      const int k = k0 + kk;
      v2f a;
      a.x = arow[k + 0];                             // folds to one b64 load, imm off
      a.y = arow[k + 1];
      const float* bb = bp + (size_t)k * NC;         // imm offsets (K,NC constexpr)
#pragma unroll
      for (int t = 0; t < NACC; ++t) {
        v2f b;
        b.x = bb[t * 16];
        b.y = bb[t * 16 + NC];
        acc[t] = __builtin_amdgcn_wmma_f32_16x16x4_f32(
            false, a, false, b, (short)0, acc[t], false, false);
      }
    }
  }

  // C/D layout: VGPR r -> M=r (lanes 0-15) / M=r+8 (lanes 16-31); N = lane&15
  const int mb = row0 + (int)((lane >> 4u) << 3u);
  float* __restrict__ c0 = C + (size_t)mb * NC + col0 + ml;
#pragma unroll
  for (int r = 0; r < 8; ++r) {
    if (mb + r < M) {
#pragma unroll
      for (int t = 0; t < NACC; ++t) c0[(size_t)r * NC + t * 16] = acc[t][r];
    }
  }
}

// ---------------------------------------------------------------------------
// Utility kernels
// ---------------------------------------------------------------------------
__global__ void fill_kernel(float* __restrict__ p, float v, size_t n) {
  size_t i = (size_t)blockIdx.x * blockDim.x + threadIdx.x;
  size_t stride = (size_t)gridDim.x * blockDim.x;
  for (; i < n; i += stride) p[i] = v;
}

__global__ void deg_count_kernel(const int* __restrict__ dst,
                                 float* __restrict__ deg, int E) {
  int i = blockIdx.x * blockDim.x + threadIdx.x;
  int stride = gridDim.x * blockDim.x;
  for (; i < E; i += stride) atomicAdd(&deg[dst[i]], 1.0f);
}

__global__ void rsqrt_kernel(const float* __restrict__ deg,
                             float* __restrict__ dinv, int n) {
  int i = blockIdx.x * blockDim.x + threadIdx.x;
  int stride = gridDim.x * blockDim.x;
  for (; i < n; i += stride) dinv[i] = rsqrtf(deg[i]);
}

// agg = xw/deg + bias  (self-loop term fused into aggregation-buffer init)
template <int F>
__global__ void init_agg_kernel(float* __restrict__ agg,
                                const float* __restrict__ xw,
                                const float* __restrict__ deg,
                                const float* __restrict__ bias, int Nn) {
  const int f = threadIdx.x;                         // blockDim.x == F
  const float bv = bias[f];
  for (int row = blockIdx.x; row < Nn; row += gridDim.x) {
    const size_t i = (size_t)row * F + f;
    agg[i] = xw[i] * (1.0f / deg[row]) + bv;
  }
}

// ---------------------------------------------------------------------------
// Edge scatter: agg[dst] += xw[src] * dinv[src]*dinv[dst]
// blockDim = (F/4, 4): edge index is wave-uniform -> scalar src/dst/coef loads;
// float4 gathers + GLOBAL_ATOMIC_ADD_F32 scatters (all L2-resident: 192MB L2).
// ---------------------------------------------------------------------------
template <int F>
__global__ void scatter_kernel(const float4* __restrict__ xw4,
                               float* __restrict__ agg,
                               const int* __restrict__ src,
                               const int* __restrict__ dst,
                               const float* __restrict__ dinv, int E) {
  constexpr int F4 = F / 4;
  const int f4 = threadIdx.x;                        // 0..F4-1
  for (int e = blockIdx.x * blockDim.y + threadIdx.y; e < E;
       e += gridDim.x * blockDim.y) {
    const int s = src[e];
    const int d = dst[e];
    const float coef = dinv[s] * dinv[d];
    const float4 v = xw4[(size_t)s * F4 + f4];
    float* dp = agg + (size_t)d * F + f4 * 4;
    atomicAdd(dp + 0, v.x * coef);
    atomicAdd(dp + 1, v.y * coef);
    atomicAdd(dp + 2, v.z * coef);
    atomicAdd(dp + 3, v.w * coef);
  }
}

// out = lrelu(agg)
template <int F>
__global__ void lrelu_kernel(const float* __restrict__ agg,
                             float* __restrict__ out, int Nn) {
  const int f = threadIdx.x;
  for (int row = blockIdx.x; row < Nn; row += gridDim.x) {
    const size_t i = (size_t)row * F + f;
    const float v = agg[i];
    out[i] = (v >= 0.0f) ? v : 0.2f * v;
  }
}

// out = (jv + bv)*0.5 + res + bias
template <int F>
__global__ void combine_kernel(const float* __restrict__ jv,
                               const float* __restrict__ bvw,
                               const float* __restrict__ res,
                               const float* __restrict__ bias,
                               float* __restrict__ out, int Nn) {
  const int f = threadIdx.x;
  const float b = bias[f];
  for (int row = blockIdx.x; row < Nn; row += gridDim.x) {
    const size_t i = (size_t)row * F + f;
    out[i] = (jv[i] + bvw[i]) * 0.5f + res[i] + b;
  }
}

// ---------------------------------------------------------------------------
// Host orchestration
// ---------------------------------------------------------------------------
extern "C" void kernel_launch(void* const* d_in, const int* in_sizes, int n_in,
                              void* d_out, int out_size, void* d_ws, size_t ws_size,
                              hipStream_t stream) {
  const int F_IN = 256, HID = 256, OUT = 128;
  const int Nn = in_sizes[0] / F_IN;      // 50000
  const int E  = in_sizes[4] / 2;         // 800000

  const float* x[4];
  for (int i = 0; i < 4; ++i) x[i] = (const float*)d_in[i];
  const int* edges[4];
  for (int i = 0; i < 4; ++i) edges[i] = (const int*)d_in[4 + i];

  float* ws     = (float*)d_ws;
  float* deg    = ws;                               // 4*Nn
  float* dinv   = deg  + (size_t)4 * Nn;            // 4*Nn
  float* bufXW  = dinv + (size_t)4 * Nn;            // Nn*HID
  float* bufH   = bufXW + (size_t)Nn * HID;         // Nn*HID
  float* bufAGG = bufH  + (size_t)Nn * HID;         // Nn*HID
  float* out    = (float*)d_out;
  const size_t NO = (size_t)Nn * OUT;
  const int mTiles = (Nn + 15) / 16;

  // ---- degree / inverse-sqrt per relation (self loop => deg starts at 1) ----
  fill_kernel<<<1024, 256, 0, stream>>>(deg, 1.0f, (size_t)4 * Nn);
  for (int r = 0; r < 4; ++r)
    deg_count_kernel<<<2048, 256, 0, stream>>>(edges[r] + E, deg + (size_t)r * Nn, E);
  rsqrt_kernel<<<1024, 256, 0, stream>>>(deg, dinv, 4 * Nn);

  // ---- one GCN layer (F_IN->HID wide, HID->OUT narrow) ----
  auto gcn_wide = [&](const float* X, const int* er, const float* degr,
                      const float* dinvr, const float* W, const float* b,
                      float* o) {
    gemm16_wmma<256, 256, 4><<<dim3(mTiles, 1), 128, 0, stream>>>(X, W, bufXW, Nn);
    init_agg_kernel<256><<<4096, 256, 0, stream>>>(bufAGG, bufXW, degr, b, Nn);
    scatter_kernel<256><<<8192, dim3(64, 4), 0, stream>>>(
        (const float4*)bufXW, bufAGG, er, er + E, dinvr, E);
    lrelu_kernel<256><<<4096, 256, 0, stream>>>(bufAGG, o, Nn);
  };
  auto gcn_narrow = [&](const float* X, const int* er, const float* degr,
                        const float* dinvr, const float* W, const float* b,
                        float* o) {
    gemm16_wmma<256, 128, 2><<<dim3(mTiles, 1), 128, 0, stream>>>(X, W, bufXW, Nn);
    init_agg_kernel<128><<<4096, 128, 0, stream>>>(bufAGG, bufXW, degr, b, Nn);
    scatter_kernel<128><<<8192, dim3(32, 4), 0, stream>>>(
        (const float4*)bufXW, bufAGG, er, er + E, dinvr, E);
    lrelu_kernel<128><<<4096, 128, 0, stream>>>(bufAGG, o, Nn);
  };

  // ---- one branch: two GCN layers, layer-2 straight into d_out section ----
  auto branch = [&](const float* X, int rel, const float* W1, const float* b1,
                    const float* W2, const float* b2, float* dst) {
    const float* degr  = deg  + (size_t)rel * Nn;
    const float* dinvr = dinv + (size_t)rel * Nn;
    gcn_wide(X,     edges[rel], degr, dinvr, W1, b1, bufH);
    gcn_narrow(bufH, edges[rel], degr, dinvr, W2, b2, dst);
  };

  // outputs: [comb_l, comb_p, jl, jp, bl, bp], each Nn x OUT
  branch(x[0], 0, (const float*)d_in[8],  (const float*)d_in[9],
                  (const float*)d_in[12], (const float*)d_in[13], out + 2 * NO); // jl
  branch(x[1], 1, (const float*)d_in[10], (const float*)d_in[11],
                  (const float*)d_in[14], (const float*)d_in[15], out + 3 * NO); // jp
  branch(x[2], 2, (const float*)d_in[16], (const float*)d_in[17],
                  (const float*)d_in[20], (const float*)d_in[21], out + 4 * NO); // bl
  branch(x[3], 3, (const float*)d_in[18], (const float*)d_in[19],
                  (const float*)d_in[22], (const float*)d_in[23], out + 5 * NO); // bp

  // ---- residual linears + fuse ----
  gemm16_wmma<256, 128, 2><<<dim3(mTiles, 1), 128, 0, stream>>>(
      x[0], (const float*)d_in[24], bufXW, Nn);
  combine_kernel<128><<<4096, 128, 0, stream>>>(
      out + 2 * NO, out + 4 * NO, bufXW, (const float*)d_in[25], out, Nn);
  gemm16_wmma<256, 128, 2><<<dim3(mTiles, 1), 128, 0, stream>>>(
      x[1], (const float*)d_in[26], bufXW, Nn);
  combine_kernel<128><<<4096, 128, 0, stream>>>(
      out + 3 * NO, out + 5 * NO, bufXW, (const float*)d_in[27], out + NO, Nn);
}